// NeuralNet_62045097558546
// MI455X (gfx1250) — compile-verified
//
#include <hip/hip_runtime.h>

typedef float v2f __attribute__((ext_vector_type(2)));
typedef float v8f __attribute__((ext_vector_type(8)));

// ---------------------------------------------------------------------------
// GEMM: C[M,Nreal] = act( A[M,K] @ B[K,Nreal] + bias ), f32 WMMA 16x16x4.
// One wave computes a 16 x (NT*16) tile; 8 waves/block -> 128 x (NT*16).
// B is staged in LDS per 64-deep K chunk (shared by all 8 waves) in a
// PAIRED-ROW layout: element (kk,c) lives at (kk/2)*PSTR + 2c + (kk&1), so a
// WMMA B fragment (rows k,k+1 of one column) is one aligned ds_load_b64.
// PSTR % 64 == 32 (dwords) => the half-wave reading pair p covers banks
// 0..31 and the half-wave reading pair p+1 covers banks 32..63: conflict-free.
// A-frag layout (ISA 7.12.2, 32-bit A 16x4): row = lane&15, vgpr j holds
// K = 2*(lane>>4)+j. C/D: vgpr i holds M = i + 8*(lane>>4), N = lane&15.
// OOB rows/cols handled by ADDRESS CLAMPING (never stored): no divergent
// guards in the hot loop.
// ---------------------------------------------------------------------------
template<int NT, bool RELU>
__global__ __launch_bounds__(256)
void gemm_bias_wmma(const float* __restrict__ A, int lda,
                    const float* __restrict__ B, int ldb,
                    const float* __restrict__ bias,
                    float* __restrict__ C, int ldc,
                    int M, int Nreal, int K)
{
    constexpr int W    = NT * 16;                       // tile width (cols)
    constexpr int KC   = 64;                            // K chunk in LDS
    constexpr int PSTR = ((2 * W) % 64 == 32) ? 2 * W   // pair stride (dwords)
                                              : 2 * W + 32 - ((2 * W) % 64);
    static_assert(PSTR % 64 == 32, "bank split requires PSTR % 64 == 32");
    __shared__ float Bs[(KC / 2) * PSTR];

    const int tid  = threadIdx.x;
    const int lane = tid & 31;
    const int wave = tid >> 5;
    const int half = lane >> 4;
    const int r    = lane & 15;

    const int rowBase = blockIdx.x * 128 + wave * 16;
    const int colBase = blockIdx.y * W;

    const int aRow = min(rowBase + r, M - 1);           // clamp, never stored

    v8f acc[NT];
#pragma unroll
    for (int t = 0; t < NT; ++t) acc[t] = (v8f){0,0,0,0,0,0,0,0};

    const float* aPtr = A + (size_t)aRow * lda + 2 * half;

    for (int k0 = 0; k0 < K; k0 += KC) {
        const int kc = min(KC, K - k0);                 // 64 or tail (mult of 4)

        // Cooperative stage of B[k0:k0+kc, colBase:colBase+W] (paired rows).
        const int elems = kc * W;
        for (int e = tid; e < elems; e += 256) {
            const int kk  = e / W;
            const int c   = e - kk * W;
            const int col = min(colBase + c, Nreal - 1);     // clamp
            Bs[(kk >> 1) * PSTR + 2 * c + (kk & 1)] =
                B[(size_t)(k0 + kk) * ldb + col];
        }
        __syncthreads();

        if (k0 + KC < K)
            __builtin_prefetch(aPtr + k0 + KC, 0, 1);        // global_prefetch_b8

        for (int kk = 0; kk < kc; kk += 4) {
            v2f a = *(const v2f*)(aPtr + k0 + kk);           // global_load_b64
            // pair index for this half-wave: kk/2 + half
            const float* bs = &Bs[(size_t)((kk >> 1) + half) * PSTR + 2 * r];
#pragma unroll
            for (int t = 0; t < NT; ++t) {
                v2f b = *(const v2f*)(bs + 32 * t);          // ds_load_b64
                acc[t] = __builtin_amdgcn_wmma_f32_16x16x4_f32(
                    false, a, false, b, (short)0, acc[t], false, false);
            }
        }
        __syncthreads();
    }

    const bool fullRows = (rowBase + 16 <= M);
#pragma unroll
    for (int t = 0; t < NT; ++t) {
        const int col = colBase + t * 16 + r;
        if (col >= Nreal) continue;
        const float bi = bias[col];
        float* cp = C + (size_t)(rowBase + 8 * half) * ldc + col;
        if (fullRows) {
#pragma unroll
            for (int i = 0; i < 8; ++i) {
                float val = acc[t][i] + bi;
                if (RELU) val = fmaxf(val, 0.0f);
                cp[(size_t)i * ldc] = val;
            }
        } else {
#pragma unroll
            for (int i = 0; i < 8; ++i) {
                if (rowBase + 8 * half + i < M) {
                    float val = acc[t][i] + bi;
                    if (RELU) val = fmaxf(val, 0.0f);
                    cp[(size_t)i * ldc] = val;
                }
            }
        }
    }
}

// ---------------------------------------------------------------------------
// Global max of cost C = max(s^2, (s-1)^2) over all activations (s >= 0, so
// float max == uint max on the bit pattern).
// ---------------------------------------------------------------------------
__global__ void zero_slot(unsigned* slot)
{
    if (threadIdx.x == 0) *slot = 0u;
}

__global__ __launch_bounds__(256)
void cost_max_reduce(const float* __restrict__ act, int ld, int n, int bs,
                     unsigned* __restrict__ slot)
{
    __shared__ float red[256];
    float m = 0.0f;
    const int total = bs * n;
    for (int i = blockIdx.x * blockDim.x + threadIdx.x; i < total;
         i += gridDim.x * blockDim.x) {
        const int row = i / n, col = i - row * n;
        const float s = act[(size_t)row * ld + col];
        m = fmaxf(m, fmaxf(s * s, (s - 1.0f) * (s - 1.0f)));
    }
    red[threadIdx.x] = m;
    __syncthreads();
    for (int s = 128; s > 0; s >>= 1) {
        if ((int)threadIdx.x < s)
            red[threadIdx.x] = fmaxf(red[threadIdx.x], red[threadIdx.x + s]);
        __syncthreads();
    }
    if (threadIdx.x == 0) atomicMax(slot, __float_as_uint(red[0]));
}

// ---------------------------------------------------------------------------
// SOFT top-k (Sinkhorn, anchors {0,1}, 50 iters) applied in place.
// One wave32 per row; 16 columns per lane kept in registers.
// ---------------------------------------------------------------------------
__global__ __launch_bounds__(256)
void sinkhorn_mask(float* __restrict__ act, int ld, int n, int bs,
                   const unsigned* __restrict__ maxSlot,
                   const int* __restrict__ sparse,
                   float eps, int kTop)
{
    const int lane = threadIdx.x & 31;
    const int row  = blockIdx.x * (blockDim.x >> 5) + (threadIdx.x >> 5);
    if (row >= bs) return;

    const float maxC   = __uint_as_float(*maxSlot);
    const float invEps = 1.0f / (eps * maxC);     // exponent = -C/(maxC*eps)
    const float sf     = (float)(*sparse);

    constexpr int TPL = 16;                        // ceil(512/32); n <= 512
    float G0[TPL], G1[TPL];
    float* rowPtr = act + (size_t)row * ld;

#pragma unroll
    for (int t = 0; t < TPL; ++t) {
        const int j = lane + 32 * t;
        if (j < n) {
            const float s = rowPtr[j];
            G0[t] = expf(-(s * s) * invEps);
            G1[t] = expf(-((s - 1.0f) * (s - 1.0f)) * invEps);
        } else {
            G0[t] = 0.0f;
            G1[t] = 0.0f;
        }
    }

    const float mu  = 1.0f / (float)n;
    const float nu0 = (float)kTop / (float)n;
    const float nu1 = (float)(n - kTop) / (float)n;
    float v0 = 0.5f, v1 = 0.5f, v0p = v0, v1p = v1;

    for (int it = 0; it < 50; ++it) {
        v0p = v0; v1p = v1;
        float s0 = 0.0f, s1 = 0.0f;
#pragma unroll
        for (int t = 0; t < TPL; ++t) {
            const float g0 = G0[t], g1 = G1[t];
            const float d = g0 * v0 + g1 * v1;
            if (d > 0.0f) {                        // invalid lanes have g0=g1=0
                const float u = mu / d;
                s0 += g0 * u;
                s1 += g1 * u;
            }
        }
#pragma unroll
        for (int off = 16; off > 0; off >>= 1) {
            s0 += __shfl_xor(s0, off, 32);
            s1 += __shfl_xor(s1, off, 32);
        }
        v0 = nu0 / s0;
        v1 = nu1 / s1;
    }

    // u_final uses v from iter 49 (v0p,v1p); gamma uses final v0.
#pragma unroll
    for (int t = 0; t < TPL; ++t) {
        const int j = lane + 32 * t;
        if (j < n) {
            const float u    = mu / (G0[t] * v0p + G1[t] * v1p);
            const float mask = (float)n * u * G0[t] * v0;
            const float blend = sf * mask + (1.0f - sf);
            rowPtr[j] = rowPtr[j] * blend;
        }
    }
}

// ---------------------------------------------------------------------------
extern "C" void kernel_launch(void* const* d_in, const int* in_sizes, int n_in,
                              void* d_out, int out_size, void* d_ws, size_t ws_size,
                              hipStream_t stream)
{
    const float* x  = (const float*)d_in[0];
    const float* W1 = (const float*)d_in[1];
    const float* b1 = (const float*)d_in[2];
    const float* W2 = (const float*)d_in[3];
    const float* b2 = (const float*)d_in[4];
    const float* W3 = (const float*)d_in[5];
    const float* b3 = (const float*)d_in[6];
    const float* W4 = (const float*)d_in[7];
    const float* b4 = (const float*)d_in[8];
    const int*   sp = (const int*)d_in[9];
    float*       out = (float*)d_out;

    const int dh   = in_sizes[2];              // 500
    const int dinF = in_sizes[1] / dh;         // 1024
    const int bs   = in_sizes[0] / dinF;       // 4096
    const int dO   = in_sizes[8];              // 10
    const int LD   = 512;                      // padded activation row stride
    const float EPS = 0.1f;
    const int   KTOP = 400;

    unsigned* slot = (unsigned*)d_ws;
    float* actA = (float*)((char*)d_ws + 256);
    float* actB = actA + (size_t)bs * LD;

    const dim3 blk(256);
    const dim3 gGemm((bs + 127) / 128, (dh + 63) / 64);   // 32 x 8
    const dim3 gOut((bs + 127) / 128, 1);
    const dim3 gSink((bs + 7) / 8);

    // Layer 1: relu(x @ W1 + b1) -> actA
    gemm_bias_wmma<4, true><<<gGemm, blk, 0, stream>>>(
        x, dinF, W1, dh, b1, actA, LD, bs, dh, dinF);
    zero_slot<<<1, 32, 0, stream>>>(slot);
    cost_max_reduce<<<256, 256, 0, stream>>>(actA, LD, dh, bs, slot);
    sinkhorn_mask<<<gSink, blk, 0, stream>>>(actA, LD, dh, bs, slot, sp, EPS, KTOP);

    // Layer 2: relu(actA @ W2 + b2) -> actB
    gemm_bias_wmma<4, true><<<gGemm, blk, 0, stream>>>(
        actA, LD, W2, dh, b2, actB, LD, bs, dh, dh);
    zero_slot<<<1, 32, 0, stream>>>(slot);
    cost_max_reduce<<<256, 256, 0, stream>>>(actB, LD, dh, bs, slot);
    sinkhorn_mask<<<gSink, blk, 0, stream>>>(actB, LD, dh, bs, slot, sp, EPS, KTOP);

    // Layer 3: relu(actB @ W3 + b3) -> actA
    gemm_bias_wmma<4, true><<<gGemm, blk, 0, stream>>>(
        actB, LD, W3, dh, b3, actA, LD, bs, dh, dh);
    zero_slot<<<1, 32, 0, stream>>>(slot);
    cost_max_reduce<<<256, 256, 0, stream>>>(actA, LD, dh, bs, slot);
    sinkhorn_mask<<<gSink, blk, 0, stream>>>(actA, LD, dh, bs, slot, sp, EPS, KTOP);

    // Layer 4: actA @ W4 + b4 -> out (no relu)
    gemm_bias_wmma<1, false><<<gOut, blk, 0, stream>>>(
        actA, LD, W4, dO, b4, out, dO, bs, dO, dh);
}